// GCN_77068893159461
// MI455X (gfx1250) — compile-verified
//
#include <hip/hip_runtime.h>
#include <hip/hip_bf16.h>

typedef float v2f __attribute__((ext_vector_type(2)));
typedef float v4f __attribute__((ext_vector_type(4)));
typedef float v8f __attribute__((ext_vector_type(8)));
typedef int   v4i __attribute__((ext_vector_type(4)));
typedef int   v8i __attribute__((ext_vector_type(8)));
typedef unsigned int u32;
typedef unsigned long long u64;
typedef u32   v4u __attribute__((ext_vector_type(4)));

#define IN_F 128
#define HID  128
#define NCOL 10
#define NCOLP 16     // padded column count for layer-2 intermediates
#define PADW 130     // LDS row stride in floats (bank-conflict-free, 8B aligned)

__device__ __forceinline__ void atomAddF(float* p, float v) {
    unsafeAtomicAdd(p, v);      // hardware global_atomic_add_f32
}

// ---------------- Tensor Data Mover staging ----------------
#if __has_builtin(__builtin_amdgcn_tensor_load_to_lds) && __has_builtin(__builtin_amdgcn_s_wait_tensorcnt)
#define USE_TDM 1
#else
#define USE_TDM 0
#endif

#if USE_TDM
__device__ __forceinline__ u32 lds_addr_of(const void* p) {
    return (u32)(uintptr_t)p;                 // AS(3) pointer -> LDS byte offset
}

// Load a (rows x row_elems) f32 tile from global into LDS.
// pad2: insert 2 dwords after every 128 dwords (LDS row stride PADW=130).
// Rows >= tensor_rows are zero-filled by TDM OOB handling.
__device__ __forceinline__ void tdm_load_f32(u32 lds_off, const void* gsrc,
                                             u32 row_elems, u32 rows,
                                             u32 tensor_rows, u32 row_stride,
                                             bool pad2) {
    u64 ga = (u64)(uintptr_t)gsrc;
    v4u g0;
    g0[0] = 1u;                                           // count=1 (user descriptor)
    g0[1] = lds_off;                                      // lds_addr (bytes)
    g0[2] = (u32)ga;                                      // global_addr[31:0]
    g0[3] = (u32)((ga >> 32) & 0x1ffffffu) | (2u << 30);  // global_addr[56:32] | type=2
    u32 w0 = (2u << 16);                                  // data_size = 4 bytes
    if (pad2) w0 |= (1u << 20)                            // pad_enable
                 |  (6u << 22)                            // pad_interval: 128 dwords
                 |  (1u << 25);                           // pad_amount: 2 dwords
    v8i g1;
    g1[0] = (int)w0;
    g1[1] = (int)((row_elems & 0xffffu) << 16);                               // tensor_dim0[15:0]
    g1[2] = (int)(((row_elems >> 16) & 0xffffu) | ((tensor_rows & 0xffffu) << 16)); // td0 hi | td1 lo
    g1[3] = (int)(((tensor_rows >> 16) & 0xffffu) | ((row_elems & 0xffffu) << 16)); // td1 hi | tile_dim0
    g1[4] = (int)(rows & 0xffffu);                                            // tile_dim1 | tile_dim2=0
    g1[5] = (int)row_stride;                                                  // tensor_dim0_stride lo32
    g1[6] = 0;
    g1[7] = 0;
    v4i z4 = {};
    v8i z8 = {};
    __builtin_amdgcn_tensor_load_to_lds(g0, g1, z4, z4, z8, 0);
}
#endif

// ---------------- degree / normalization ----------------
__global__ __launch_bounds__(256) void k_deg_init(float* deg, int n) {
    int i = blockIdx.x * blockDim.x + threadIdx.x;
    if (i < n) deg[i] = 1.0f;                 // self-loop contribution
}

__global__ __launch_bounds__(256) void k_deg_count(float* deg, const int* __restrict__ dst, int e) {
    int i = blockIdx.x * blockDim.x + threadIdx.x;
    if (i < e) atomAddF(&deg[dst[i]], 1.0f);
}

__global__ __launch_bounds__(256) void k_rsqrt(float* deg, int n) {
    int i = blockIdx.x * blockDim.x + threadIdx.x;
    if (i < n) deg[i] = rsqrtf(deg[i]);
}

// ---------------- weight prep: W1T[c][k] (128xPADW), W2T[c][k] (16xPADW, zero-padded) ----------------
__global__ __launch_bounds__(256) void k_prep_w(const float* __restrict__ W1,
                                                const float* __restrict__ W2,
                                                float* __restrict__ W1T,
                                                float* __restrict__ W2T) {
    int idx = blockIdx.x * blockDim.x + threadIdx.x;
    if (idx < HID * PADW) {
        int c = idx / PADW, k = idx % PADW;
        W1T[idx] = (k < IN_F) ? W1[k * HID + c] : 0.0f;
    }
    if (idx < NCOLP * PADW) {
        int c = idx / PADW, k = idx % PADW;
        W2T[idx] = (k < HID && c < NCOL) ? W2[k * NCOL + c] : 0.0f;
    }
}

// ---------------- GEMM1: h1[N,128] = x[N,128] @ W1[128,128] (WMMA f32 16x16x4) ----------------
__global__ __launch_bounds__(256) void k_gemm1(const float* __restrict__ x,
                                               const float* __restrict__ W1T,
                                               float* __restrict__ h1, int n) {
    __shared__ float xs[16 * PADW];           // padded x tile
    __shared__ float wls[HID * PADW];         // padded transposed W1
    const int row0 = blockIdx.x * 16;
#if USE_TDM
    if (threadIdx.x == 0) {
        tdm_load_f32(lds_addr_of(xs),  x + (size_t)row0 * IN_F,
                     IN_F, 16, (u32)(n - row0), IN_F, true);
        tdm_load_f32(lds_addr_of(wls), W1T, HID * PADW, 1, 1, HID * PADW, false);
        __builtin_amdgcn_s_wait_tensorcnt(0);
    }
#else
    for (int it = 0; it < 8; ++it) {          // 16*128 elems
        int idx = it * 256 + threadIdx.x;
        int r = idx >> 7, c = idx & 127;
        xs[r * PADW + c] = (row0 + r < n) ? x[(size_t)(row0 + r) * IN_F + c] : 0.0f;
    }
    for (int it = 0; it < (HID * PADW) / 256; ++it) {   // 16640/256 = 65
        int idx = it * 256 + threadIdx.x;
        wls[idx] = W1T[idx];
    }
#endif
    __syncthreads();

    const int t    = threadIdx.x;
    const int wave = t >> 5;
    const int lane = t & 31;
    const int col0 = wave * 16;               // 8 waves cover 128 cols
    const int m    = lane & 15;
    const int kh   = lane >> 4;

    v8f acc = {};
    #pragma unroll
    for (int k0 = 0; k0 < IN_F; k0 += 4) {
        const int ka = k0 + 2 * kh;
        v2f a = *(const v2f*)&xs[m * PADW + ka];
        v2f b = *(const v2f*)&wls[(col0 + m) * PADW + ka];
        acc = __builtin_amdgcn_wmma_f32_16x16x4_f32(false, a, false, b,
                                                    (short)0, acc, false, false);
    }
    #pragma unroll
    for (int r = 0; r < 8; ++r) {
        int row = row0 + r + 8 * kh;
        if (row < n) h1[(size_t)row * HID + col0 + m] = acc[r];
    }
}

// ---------------- aggregation layer 1 ----------------
__global__ __launch_bounds__(256) void k_selfinit1(const float* __restrict__ h1,
                                                   const float* __restrict__ isq,
                                                   float* __restrict__ agg1, int n) {
    int i = blockIdx.x * blockDim.x + threadIdx.x;    // over n*32 float4s
    if (i >= n * 32) return;
    int node = i >> 5;
    float w = isq[node]; w *= w;
    v4f v = ((const v4f*)h1)[i];
    v *= w;
    ((v4f*)agg1)[i] = v;
}

__global__ __launch_bounds__(256) void k_scatter1(const float* __restrict__ h1,
                                                  float* __restrict__ agg1,
                                                  const int* __restrict__ src,
                                                  const int* __restrict__ dst,
                                                  const float* __restrict__ isq, int e) {
    int gw   = (blockIdx.x * blockDim.x + threadIdx.x) >> 5;  // edge id (wave-uniform)
    int lane = threadIdx.x & 31;
    if (gw >= e) return;
    int s = src[gw], d = dst[gw];
    float w = isq[s] * isq[d];
    v4f hv = ((const v4f*)(h1 + (size_t)s * HID))[lane];
    float* out = agg1 + (size_t)d * HID + lane * 4;
    atomAddF(out + 0, hv[0] * w);
    atomAddF(out + 1, hv[1] * w);
    atomAddF(out + 2, hv[2] * w);
    atomAddF(out + 3, hv[3] * w);
}

__global__ __launch_bounds__(256) void k_bias_relu(float* __restrict__ agg1,
                                                   const float* __restrict__ b1, int n) {
    int i = blockIdx.x * blockDim.x + threadIdx.x;    // over n*32 float4s
    if (i >= n * 32) return;
    int cb = (i & 31) * 4;
    v4f v = ((const v4f*)agg1)[i];
    v[0] = fmaxf(v[0] + b1[cb + 0], 0.0f);
    v[1] = fmaxf(v[1] + b1[cb + 1], 0.0f);
    v[2] = fmaxf(v[2] + b1[cb + 2], 0.0f);
    v[3] = fmaxf(v[3] + b1[cb + 3], 0.0f);
    ((v4f*)agg1)[i] = v;
}

// ---------------- GEMM2: h2[N,16] = agg1[N,128] @ W2pad[128,16] ----------------
__global__ __launch_bounds__(256) void k_gemm2(const float* __restrict__ agg1,
                                               const float* __restrict__ W2T,
                                               float* __restrict__ h2, int n) {
    __shared__ float as_[128 * PADW];         // padded 128-row tile of agg1 (~65 KB)
    __shared__ float w2s[NCOLP * PADW];       // padded transposed W2
    const int rowblk = blockIdx.x * 128;
#if USE_TDM
    if (threadIdx.x == 0) {
        tdm_load_f32(lds_addr_of(as_), agg1 + (size_t)rowblk * HID,
                     HID, 128, (u32)(n - rowblk), HID, true);
        tdm_load_f32(lds_addr_of(w2s), W2T, NCOLP * PADW, 1, 1, NCOLP * PADW, false);
        __builtin_amdgcn_s_wait_tensorcnt(0);
    }
#else
    for (int it = 0; it < 64; ++it) {         // 128*128 elems
        int idx = it * 256 + threadIdx.x;
        int r = idx >> 7, c = idx & 127;
        as_[r * PADW + c] = (rowblk + r < n) ? agg1[(size_t)(rowblk + r) * HID + c] : 0.0f;
    }
    for (int it = 0; it < 9; ++it) {
        int idx = it * 256 + threadIdx.x;
        if (idx < NCOLP * PADW) w2s[idx] = W2T[idx];
    }
#endif
    __syncthreads();

    const int t    = threadIdx.x;
    const int wave = t >> 5;                  // each wave: 16 rows, 16 (padded) cols
    const int lane = t & 31;
    const int m    = lane & 15;
    const int kh   = lane >> 4;
    const int row0 = rowblk + wave * 16;

    v8f acc = {};
    #pragma unroll
    for (int k0 = 0; k0 < HID; k0 += 4) {
        const int ka = k0 + 2 * kh;
        v2f a = *(const v2f*)&as_[(wave * 16 + m) * PADW + ka];
        v2f b = *(const v2f*)&w2s[m * PADW + ka];
        acc = __builtin_amdgcn_wmma_f32_16x16x4_f32(false, a, false, b,
                                                    (short)0, acc, false, false);
    }
    #pragma unroll
    for (int r = 0; r < 8; ++r) {
        int row = row0 + r + 8 * kh;
        if (row < n) h2[(size_t)row * NCOLP + m] = acc[r];   // padded cols exact 0
    }
}

// ---------------- aggregation layer 2 ----------------
__global__ __launch_bounds__(256) void k_selfinit2(const float* __restrict__ h2,
                                                   const float* __restrict__ isq,
                                                   float* __restrict__ agg2, int n) {
    int i = blockIdx.x * blockDim.x + threadIdx.x;    // over n*16 floats
    if (i >= n * NCOLP) return;
    int node = i >> 4;
    float w = isq[node]; w *= w;
    agg2[i] = h2[i] * w;
}

__global__ __launch_bounds__(256) void k_scatter2(const float* __restrict__ h2,
                                                  float* __restrict__ agg2,
                                                  const int* __restrict__ src,
                                                  const int* __restrict__ dst,
                                                  const float* __restrict__ isq, int e) {
    int gid = blockIdx.x * blockDim.x + threadIdx.x;
    int edge = gid >> 4;
    int c    = gid & 15;
    if (edge >= e || c >= NCOL) return;
    int s = src[edge], d = dst[edge];
    float w = isq[s] * isq[d];
    atomAddF(&agg2[(size_t)d * NCOLP + c], h2[(size_t)s * NCOLP + c] * w);
}

// ---------------- softmax(agg2 + b2) -> out[N,10] ----------------
__global__ __launch_bounds__(256) void k_softmax(const float* __restrict__ agg2,
                                                 const float* __restrict__ b2,
                                                 float* __restrict__ out, int n) {
    int i = blockIdx.x * blockDim.x + threadIdx.x;
    if (i >= n) return;
    float v[NCOL];
    float mx = -3.402823466e38f;
    #pragma unroll
    for (int c = 0; c < NCOL; ++c) {
        v[c] = agg2[(size_t)i * NCOLP + c] + b2[c];
        mx = fmaxf(mx, v[c]);
    }
    float sum = 0.0f;
    #pragma unroll
    for (int c = 0; c < NCOL; ++c) {
        v[c] = __expf(v[c] - mx);
        sum += v[c];
    }
    float inv = 1.0f / sum;
    #pragma unroll
    for (int c = 0; c < NCOL; ++c)
        out[(size_t)i * NCOL + c] = v[c] * inv;
}

extern "C" void kernel_launch(void* const* d_in, const int* in_sizes, int n_in,
                              void* d_out, int out_size, void* d_ws, size_t ws_size,
                              hipStream_t stream) {
    const float* x  = (const float*)d_in[0];
    const int*   ei = (const int*)d_in[1];
    const float* W1 = (const float*)d_in[2];
    const float* b1 = (const float*)d_in[3];
    const float* W2 = (const float*)d_in[4];
    const float* b2 = (const float*)d_in[5];

    const int N = in_sizes[0] / IN_F;
    const int E = in_sizes[1] / 2;
    const int* src = ei;        // edge_index[0]
    const int* dst = ei + E;    // edge_index[1]

    float* ws = (float*)d_ws;
    size_t off = 0;
    float* isq  = ws + off; off += (size_t)((N + 255) & ~255);
    float* h1   = ws + off; off += (size_t)N * HID;
    float* agg1 = ws + off; off += (size_t)N * HID;
    float* h2   = ws + off; off += (size_t)N * NCOLP;
    float* agg2 = ws + off; off += (size_t)N * NCOLP;
    float* W1T  = ws + off; off += (size_t)HID * PADW;
    float* W2T  = ws + off; off += (size_t)NCOLP * PADW;
    float* out  = (float*)d_out;

    // degree -> isq ; weight transposes
    k_deg_init <<<(N + 255) / 256, 256, 0, stream>>>(isq, N);
    k_deg_count<<<(E + 255) / 256, 256, 0, stream>>>(isq, dst, E);
    k_rsqrt    <<<(N + 255) / 256, 256, 0, stream>>>(isq, N);
    k_prep_w   <<<(HID * PADW + 255) / 256, 256, 0, stream>>>(W1, W2, W1T, W2T);

    // layer 1
    k_gemm1    <<<(N + 15) / 16, 256, 0, stream>>>(x, W1T, h1, N);
    k_selfinit1<<<((size_t)N * 32 + 255) / 256, 256, 0, stream>>>(h1, isq, agg1, N);
    k_scatter1 <<<((size_t)E * 32 + 255) / 256, 256, 0, stream>>>(h1, agg1, src, dst, isq, E);
    k_bias_relu<<<((size_t)N * 32 + 255) / 256, 256, 0, stream>>>(agg1, b1, N);

    // layer 2
    k_gemm2    <<<(N + 127) / 128, 256, 0, stream>>>(agg1, W2T, h2, N);
    k_selfinit2<<<((size_t)N * NCOLP + 255) / 256, 256, 0, stream>>>(h2, isq, agg2, N);
    k_scatter2 <<<((size_t)E * 16 + 255) / 256, 256, 0, stream>>>(h2, agg2, src, dst, isq, E);

    // softmax
    k_softmax  <<<(N + 255) / 256, 256, 0, stream>>>(agg2, b2, out, N);
}